// SpectralConv2d_42382737277126
// MI455X (gfx1250) — compile-verified
//
#include <hip/hip_runtime.h>
#include <math.h>

// SpectralConv2d for MI455X (gfx1250): truncated-DFT formulation.
// Every FFT stage becomes a dense GEMM vs a twiddle matrix, executed with
// V_WMMA_F32_16X16X4_F32 (f32 WMMA keeps DFT precision; problem is
// memory-bound at ~190MB / 23.3TB/s so low-precision buys nothing).

static constexpr int kH = 512;
static constexpr int kW = 512;
static constexpr int kC = 64;    // in channels
static constexpr int kO = 64;    // out channels
static constexpr int kM1 = 32;   // kx modes
static constexpr int kM2 = 32;   // ky modes
static constexpr int kModes = kM1 * kM2;       // 1024
static constexpr float kInvHW = 1.0f / (512.0f * 512.0f);

typedef __attribute__((ext_vector_type(2))) float v2f;
typedef __attribute__((ext_vector_type(8))) float v8f;

// D(16x16,f32) = A(16x4) * B(4x16) + C
__device__ inline v8f wmma4(v2f a, v2f b, v8f c) {
  return __builtin_amdgcn_wmma_f32_16x16x4_f32(
      /*neg_a=*/false, a, /*neg_b=*/false, b,
      /*c_mod=*/(short)0, c, /*reuse_a=*/false, /*reuse_b=*/false);
}

// ---------------------------------------------------------------------------
// Twiddle tables (recomputed every call; deterministic, ~130K floats).
//  Bw [512][64] : n<32 ->  cos(2pi*w*n/512),  n>=32 -> -sin(...)      (fwd W)
//  Dm [64][512] : m<32 ->  cos(2pi*h*m/512),  m>=32 -> +sin(...)      (fwd H)
//  E2 [512][64] : k<32 ->  cos(2pi*h*k/512),  k>=32 -> +sin(...)      (inv H)
//  G  [64][512] : k<32 ->  c_k/HW * cos(2pi*w*k/512), k>=32 -> -c_k/HW*sin
//                 with c_0 = 1, c_k = 2 (Hermitian fold of rfft axis)
// ---------------------------------------------------------------------------
__global__ __launch_bounds__(256)
void build_twiddles(float* __restrict__ Bw, float* __restrict__ Dm,
                    float* __restrict__ E2, float* __restrict__ G) {
  const int i = blockIdx.x * 256 + threadIdx.x;      // 0..32767
  const float step = 6.2831853071795864769f / 512.0f;
  { // Bw: row=w (i>>6), col=n (i&63)
    int r = i >> 6, c = i & 63, k = c & 31;
    float th = ((r * k) & 511) * step;
    Bw[i] = (c < 32) ? cosf(th) : -sinf(th);
  }
  { // Dm: row=m (i>>9), col=h (i&511)
    int r = i >> 9, hc = i & 511, k = r & 31;
    float th = ((hc * k) & 511) * step;
    Dm[i] = (r < 32) ? cosf(th) : sinf(th);
  }
  { // E2: row=h (i>>6), col=k (i&63)
    int hr = i >> 6, c = i & 63, k = c & 31;
    float th = ((hr * k) & 511) * step;
    E2[i] = (c < 32) ? cosf(th) : sinf(th);
  }
  { // G: row=k (i>>9), col=w (i&511)
    int r = i >> 9, w = i & 511, k = r & 31;
    float th = ((w * k) & 511) * step;
    float s = ((k == 0) ? 1.0f : 2.0f) * kInvHW;
    G[i] = (r < 32) ? s * cosf(th) : -s * sinf(th);
  }
}

// ---------------------------------------------------------------------------
// Stage A: X1(32768x64) = x(32768x512) * Bw(512x64).
// Block = 256 thr (8 waves); each block does a 32-row strip, each wave one
// 16x16 tile (2 M-tiles x 4 N-tiles). Bw staged in 128KB LDS.
// ---------------------------------------------------------------------------
__global__ __launch_bounds__(256)
void row_dft(const float* __restrict__ x, const float* __restrict__ Bw,
             float* __restrict__ X1) {
  extern __shared__ float Bs[];                  // 512*64 floats = 128KB
  for (int i = threadIdx.x; i < 512 * 64; i += 256) Bs[i] = Bw[i];
  __syncthreads();

  const int wave = threadIdx.x >> 5, lane = threadIdx.x & 31;
  const int mt = wave >> 2, nt = wave & 3;
  const int hh = lane >> 4, lm = lane & 15;
  const int rowBase = blockIdx.x * 32 + mt * 16;
  const float* arow = x + (rowBase + lm) * 512;
  const int col = nt * 16 + lm;

  v8f acc = {};
  for (int k0 = 0; k0 < 512; k0 += 4) {
    const int ka = k0 + 2 * hh;
    v2f a; a.x = arow[ka]; a.y = arow[ka + 1];
    v2f b; b.x = Bs[ka * 64 + col]; b.y = Bs[(ka + 1) * 64 + col];
    acc = wmma4(a, b, acc);
  }
  for (int r = 0; r < 8; ++r)
    X1[(rowBase + r + 8 * hh) * 64 + col] = acc[r];
}

// ---------------------------------------------------------------------------
// Stage B: per channel c: P(64x64) = Dm(64x512) * X1_c(512x64); then combine
//   X2r[kx,ky] = P[kx,ky]     + P[32+kx,32+ky]
//   X2i[kx,ky] = P[kx,32+ky]  - P[32+kx,ky]         ( e^{-i\theta} mix )
// Block = 512 thr (16 waves), one wave per 16x16 tile of P.
// ---------------------------------------------------------------------------
__global__ __launch_bounds__(512)
void col_dft(const float* __restrict__ X1, const float* __restrict__ Dm,
             float* __restrict__ X2r, float* __restrict__ X2i) {
  __shared__ float P[64 * 64];                   // 16KB
  const int c = blockIdx.x;
  const int wave = threadIdx.x >> 5, lane = threadIdx.x & 31;
  const int mt = wave >> 2, nt = wave & 3;
  const int hh = lane >> 4, lm = lane & 15;
  const int arow = mt * 16 + lm;
  const int col = nt * 16 + lm;
  const float* B = X1 + c * 512 * 64;

  v8f acc = {};
  for (int k0 = 0; k0 < 512; k0 += 4) {
    const int ka = k0 + 2 * hh;
    v2f a; a.x = Dm[arow * 512 + ka]; a.y = Dm[arow * 512 + ka + 1];
    v2f b; b.x = B[ka * 64 + col];    b.y = B[(ka + 1) * 64 + col];
    acc = wmma4(a, b, acc);
  }
  for (int r = 0; r < 8; ++r)
    P[(mt * 16 + r + 8 * hh) * 64 + col] = acc[r];
  __syncthreads();

  for (int idx = threadIdx.x; idx < kModes; idx += 512) {
    const int kx = idx >> 5, ky = idx & 31;
    X2r[c * kModes + idx] = P[kx * 64 + ky]        + P[(32 + kx) * 64 + 32 + ky];
    X2i[c * kModes + idx] = P[kx * 64 + 32 + ky]   - P[(32 + kx) * 64 + ky];
  }
}

// ---------------------------------------------------------------------------
// Stage C: per-mode complex channel mix (weights differ per mode -> matvecs,
// not WMMA). Parallelized so weight reads are coalesced along the mode axis:
// grid = (4, 64 out-ch), thread = mode. 32MB weight traffic, read once.
// ---------------------------------------------------------------------------
__global__ __launch_bounds__(256)
void mode_mix(const float* __restrict__ X2r, const float* __restrict__ X2i,
              const float* __restrict__ Wr, const float* __restrict__ Wi,
              float* __restrict__ Fr, float* __restrict__ Fi) {
  const int o = blockIdx.y;
  const int m = blockIdx.x * 256 + threadIdx.x;  // 0..1023
  float ar = 0.0f, ai = 0.0f;
  for (int c = 0; c < kC; ++c) {
    const float xr = X2r[c * kModes + m];
    const float xi = X2i[c * kModes + m];
    const int wofs = ((c * kO + o) << 10) + m;
    const float wr = Wr[wofs], wi = Wi[wofs];
    ar = fmaf(xr, wr, fmaf(-xi, wi, ar));
    ai = fmaf(xr, wi, fmaf(xi, wr, ai));
  }
  Fr[(o << 10) + m] = ar;
  Fi[(o << 10) + m] = ai;
}

// ---------------------------------------------------------------------------
// Stage D: per out-channel o: T_o(512x64) = E2(512x64) * Bm(64x64), where
//   Bm = [ Fr  Fi ; -Fi  Fr ]  built in LDS  ( e^{+i\theta} expansion )
// T columns: [Tr(32) | Ti(32)]. Block = 512 thr (16 waves), 8 tiles/wave.
// ---------------------------------------------------------------------------
__global__ __launch_bounds__(512)
void inv_col(const float* __restrict__ Fr, const float* __restrict__ Fi,
             const float* __restrict__ E2, float* __restrict__ T) {
  __shared__ float Bs[64 * 64];                  // 16KB
  const int o = blockIdx.x;
  for (int idx = threadIdx.x; idx < 64 * 64; idx += 512) {
    const int r = idx >> 6, n = idx & 63;
    const int fofs = (o << 10) + (r & 31) * 32 + (n & 31);
    float v;
    if (r < 32) v = (n < 32) ?  Fr[fofs] : Fi[fofs];
    else        v = (n < 32) ? -Fi[fofs] : Fr[fofs];
    Bs[idx] = v;
  }
  __syncthreads();

  const int wave = threadIdx.x >> 5, lane = threadIdx.x & 31;
  const int hh = lane >> 4, lm = lane & 15;
  for (int mt = wave; mt < 32; mt += 16) {
    const int rowBase = mt * 16;
    const float* arow = E2 + (rowBase + lm) * 64;
    for (int nt = 0; nt < 4; ++nt) {
      const int col = nt * 16 + lm;
      v8f acc = {};
      for (int k0 = 0; k0 < 64; k0 += 4) {
        const int ka = k0 + 2 * hh;
        v2f a; a.x = arow[ka];            a.y = arow[ka + 1];
        v2f b; b.x = Bs[ka * 64 + col];   b.y = Bs[(ka + 1) * 64 + col];
        acc = wmma4(a, b, acc);
      }
      for (int r = 0; r < 8; ++r)
        T[(o * 512 + rowBase + r + 8 * hh) * 64 + col] = acc[r];
    }
  }
}

// ---------------------------------------------------------------------------
// Stage E: y(32768x512) = T(32768x64) * G(64x512).  G (128KB) in LDS.
// Block = 256 thr (8 waves) per 16-row strip; each wave 4 N-tiles.
// ---------------------------------------------------------------------------
__global__ __launch_bounds__(256)
void inv_row(const float* __restrict__ T, const float* __restrict__ G,
             float* __restrict__ y) {
  extern __shared__ float Gs[];                  // 64*512 floats = 128KB
  for (int i = threadIdx.x; i < 64 * 512; i += 256) Gs[i] = G[i];
  __syncthreads();

  const int wave = threadIdx.x >> 5, lane = threadIdx.x & 31;
  const int hh = lane >> 4, lm = lane & 15;
  const int rb = blockIdx.x * 16;
  const float* arow = T + (rb + lm) * 64;

  for (int nt = wave; nt < 32; nt += 8) {
    const int col = nt * 16 + lm;
    v8f acc = {};
    for (int k0 = 0; k0 < 64; k0 += 4) {
      const int ka = k0 + 2 * hh;
      v2f a; a.x = arow[ka];             a.y = arow[ka + 1];
      v2f b; b.x = Gs[ka * 512 + col];   b.y = Gs[(ka + 1) * 512 + col];
      acc = wmma4(a, b, acc);
    }
    for (int r = 0; r < 8; ++r)
      y[(rb + r + 8 * hh) * 512 + col] = acc[r];
  }
}

// ---------------------------------------------------------------------------
extern "C" void kernel_launch(void* const* d_in, const int* in_sizes, int n_in,
                              void* d_out, int out_size, void* d_ws, size_t ws_size,
                              hipStream_t stream) {
  const float* x  = (const float*)d_in[0];   // (64,512,512)
  const float* Wr = (const float*)d_in[1];   // (64,64,32,32)
  const float* Wi = (const float*)d_in[2];   // (64,64,32,32)
  float* y = (float*)d_out;                  // (64,512,512)

  // Workspace map (floats). Total 4,587,520 floats = ~17.5 MB.
  float* ws  = (float*)d_ws;
  float* Bw  = ws;                   // 32768
  float* Dm  = Bw  + 32768;          // 32768
  float* E2  = Dm  + 32768;          // 32768
  float* G   = E2  + 32768;          // 32768
  float* X1  = G   + 32768;          // 32768*64 = 2,097,152
  float* X2r = X1  + 2097152;        // 64*1024
  float* X2i = X2r + 65536;          // 64*1024
  float* Fr  = X2i + 65536;          // 64*1024
  float* Fi  = Fr  + 65536;          // 64*1024
  float* T   = Fi  + 65536;          // 32768*64 = 2,097,152

  build_twiddles<<<128, 256, 0, stream>>>(Bw, Dm, E2, G);
  row_dft<<<1024, 256, 512 * 64 * sizeof(float), stream>>>(x, Bw, X1);
  col_dft<<<64, 512, 0, stream>>>(X1, Dm, X2r, X2i);
  mode_mix<<<dim3(4, 64), 256, 0, stream>>>(X2r, X2i, Wr, Wi, Fr, Fi);
  inv_col<<<64, 512, 0, stream>>>(Fr, Fi, E2, T);
  inv_row<<<2048, 256, 64 * 512 * sizeof(float), stream>>>(T, G, y);
}